// OptimalTransportResampler_84327387890382
// MI455X (gfx1250) — compile-verified
//
#include <hip/hip_runtime.h>
#include <hip/hip_bf16.h>

typedef float v2f __attribute__((ext_vector_type(2)));
typedef float v8f __attribute__((ext_vector_type(8)));

#define NP   2048
#define DIM  16
#define NB   4
#define EPS_TGT 0.01f
#define INV_REG 100.0f           // 1/REG
#define MIN_UPD 0.01f
#define DECAY   0.9f
#define MAX_IT  50
#define LOGB  (-7.6246189861593985f)   // -log(2048)

// ---- monotone float<->uint mapping (for atomic min/max over signed floats) ----
__device__ __forceinline__ unsigned fmono(float x) {
  unsigned u = __float_as_uint(x);
  return (u & 0x80000000u) ? ~u : (u | 0x80000000u);
}
__device__ __forceinline__ float funmono(unsigned u) {
  return __uint_as_float((u & 0x80000000u) ? (u & 0x7FFFFFFFu) : ~u);
}

// =====================================================================
// Kernel 1: preprocess. One block per batch.
//   mean/std per dim -> diameter -> scale; sx = (state-mean)/scale;
//   sq[i] = |sx_i|^2 ; extent -> eps0 ; zero f,g,upd,done.
// =====================================================================
__global__ __launch_bounds__(256) void ot_prep(
    const float* __restrict__ state,
    float* __restrict__ sx, float* __restrict__ sq,
    float* __restrict__ f, float* __restrict__ g,
    float* __restrict__ eps, int* __restrict__ done,
    unsigned int* __restrict__ upd)
{
  int b = blockIdx.x;
  int t = threadIdx.x;
  __shared__ float ssum[DIM], ssq[DIM];
  __shared__ unsigned smin, smax;
  __shared__ float s_mean[DIM];
  __shared__ float s_invscale;

  if (t < DIM) { ssum[t] = 0.f; ssq[t] = 0.f; }
  if (t == 0)  { smin = 0xFFFFFFFFu; smax = 0u; }
  __syncthreads();

  float ls[DIM], ls2[DIM];
#pragma unroll
  for (int d = 0; d < DIM; ++d) { ls[d] = 0.f; ls2[d] = 0.f; }
  for (int n = t; n < NP; n += 256) {
    const float* p = state + ((size_t)(b * NP + n)) * DIM;
#pragma unroll
    for (int d = 0; d < DIM; ++d) { float v = p[d]; ls[d] += v; ls2[d] += v * v; }
  }
#pragma unroll
  for (int d = 0; d < DIM; ++d) {
    atomicAdd(&ssum[d], ls[d]);
    atomicAdd(&ssq[d],  ls2[d]);
  }
  __syncthreads();

  if (t == 0) {
    float diam = 0.f;
#pragma unroll
    for (int d = 0; d < DIM; ++d) {
      float mean = ssum[d] * (1.0f / NP);
      float var  = fmaxf(ssq[d] * (1.0f / NP) - mean * mean, 0.f);
      diam = fmaxf(diam, sqrtf(var));
      s_mean[d] = mean;
    }
    if (diam == 0.f) diam = 1.f;
    s_invscale = 1.0f / (diam * 4.0f);   // sqrt(D)=4
  }
  __syncthreads();

  unsigned lmin = 0xFFFFFFFFu, lmax = 0u;
  float invs = s_invscale;
  for (int n = t; n < NP; n += 256) {
    const float* p = state + ((size_t)(b * NP + n)) * DIM;
    float* q = sx + ((size_t)(b * NP + n)) * DIM;
    float s = 0.f;
#pragma unroll
    for (int d = 0; d < DIM; ++d) {
      float v = (p[d] - s_mean[d]) * invs;
      q[d] = v;
      s += v * v;
      unsigned u = fmono(v);
      lmin = (u < lmin) ? u : lmin;
      lmax = (u > lmax) ? u : lmax;
    }
    sq[b * NP + n] = s;
    f[b * NP + n] = 0.f;
    g[b * NP + n] = 0.f;
  }
  atomicMin(&smin, lmin);
  atomicMax(&smax, lmax);
  __syncthreads();

  if (t == 0) {
    eps[b]  = funmono(smax) - funmono(smin);   // extent = eps0
    done[b] = 0;
    upd[b]  = 0;
  }
}

// =====================================================================
// Kernel 2: one Sinkhorn half-step (symmetric cost -> same kernel for f & g).
//   out[i] = sq[i] - eps * LSE_j( c_j + (2/eps) * G[i,j] ),
//   c_j = prior[j] + (dual[j] - sq[j]) / eps,  G = sx * sx^T via WMMA f32.
// One wave per 16-row tile; 128 col-chunks of 16; online log-sum-exp.
// =====================================================================
__global__ __launch_bounds__(256) void sinkhorn_half(
    const float* __restrict__ sx, const float* __restrict__ sq,
    const float* __restrict__ dual,        // g for f-step, f for g-step
    float* __restrict__ outv,              // f for f-step, g for g-step
    const float* __restrict__ prior,       // nullptr -> constant log_b
    float prior_const,
    const float* __restrict__ eps_arr,
    const int* __restrict__ done_arr,
    unsigned int* __restrict__ upd_arr)
{
  int wave = (int)((blockIdx.x * blockDim.x + threadIdx.x) >> 5);
  int lane = (int)(threadIdx.x & 31);
  int b = wave >> 7;                 // 128 row-tiles per batch
  int rowbase = (wave & 127) << 4;
  if (done_arr[b]) return;           // uniform per block (8 tiles/block, 128%8==0)

  float eps = eps_arr[b];
  float inv_eps = 1.0f / eps;
  float two_inv_eps = 2.0f * inv_eps;

  const float* sxb = sx + (size_t)b * NP * DIM;
  const float* sqb = sq + (size_t)b * NP;
  const float* dub = dual + (size_t)b * NP;
  const float* prb = prior ? (prior + (size_t)b * NP) : nullptr;

  int mloc = lane & 15;
  int koff = (lane < 16) ? 0 : 2;    // 32-bit A/B layout: VGPR {K, K+1}, halves {0,2}

  // A tile: rows [rowbase, rowbase+16), all 16 K-dims -> 4 x v2f (loop invariant)
  v2f a[4];
#pragma unroll
  for (int kk = 0; kk < 4; ++kk) {
    const float* p = sxb + (size_t)(rowbase + mloc) * DIM + 4 * kk + koff;
    a[kk].x = p[0];
    a[kk].y = p[1];
  }

  float mrun[8], srun[8];
#pragma unroll
  for (int v = 0; v < 8; ++v) { mrun[v] = -__builtin_inff(); srun[v] = 0.f; }

  for (int cb = 0; cb < NP; cb += 16) {
    int j = cb + mloc;
    float cc = (prb ? prb[j] : prior_const) + (dub[j] - sqb[j]) * inv_eps;

    // B tile: B[k][n] = sx[col n][k] -> same indexing pattern as A with col base
    v2f bt[4];
#pragma unroll
    for (int kk = 0; kk < 4; ++kk) {
      const float* p = sxb + (size_t)(cb + mloc) * DIM + 4 * kk + koff;
      bt[kk].x = p[0];
      bt[kk].y = p[1];
    }

    v8f c = {0.f, 0.f, 0.f, 0.f, 0.f, 0.f, 0.f, 0.f};
#pragma unroll
    for (int kk = 0; kk < 4; ++kk) {
      c = __builtin_amdgcn_wmma_f32_16x16x4_f32(
          /*neg_a=*/false, a[kk], /*neg_b=*/false, bt[kk],
          /*c_mod=*/(short)0, c, /*reuse_a=*/false, /*reuse_b=*/false);
    }

    // C layout: lane L holds col (L&15); VGPR slot v holds row v (+8 for upper half).
#pragma unroll
    for (int v = 0; v < 8; ++v) {
      float t  = cc + two_inv_eps * c[v];
      float mo = mrun[v];
      float mn = fmaxf(mo, t);
      srun[v] = srun[v] * __expf(mo - mn) + __expf(t - mn);
      mrun[v] = mn;
    }
  }

  // merge the 16 lane-local (max,sum) pairs inside each 16-lane half
#pragma unroll
  for (int off = 1; off <= 8; off <<= 1) {
#pragma unroll
    for (int v = 0; v < 8; ++v) {
      float mo = __shfl_xor(mrun[v], off, 32);
      float so = __shfl_xor(srun[v], off, 32);
      float mn = fmaxf(mrun[v], mo);
      srun[v] = srun[v] * __expf(mrun[v] - mn) + so * __expf(mo - mn);
      mrun[v] = mn;
    }
  }

  if (mloc == 0) {
    int rbase = rowbase + ((lane < 16) ? 0 : 8);
#pragma unroll
    for (int v = 0; v < 8; ++v) {
      int i = rbase + v;
      float val = sqb[i] - eps * (mrun[v] + __logf(srun[v]));
      float old = outv[b * NP + i];
      outv[b * NP + i] = val;
      atomicMax(&upd_arr[b], __float_as_uint(fabsf(val - old)));  // >=0: bits monotone
    }
  }
}

// =====================================================================
// Kernel 3: per-iteration convergence check + eps decay (per batch).
// =====================================================================
__global__ void sinkhorn_update(float* __restrict__ eps, int* __restrict__ done,
                                unsigned int* __restrict__ upd)
{
  int b = threadIdx.x;
  if (b >= NB) return;
  if (!done[b]) {
    float u = __uint_as_float(upd[b]);
    int conv = (u < MIN_UPD) && (eps[b] <= EPS_TGT);   // uses eps2 (pre-decay)
    eps[b] = fmaxf(eps[b] * DECAY, EPS_TGT);
    if (conv) done[b] = 1;
  }
  upd[b] = 0;
}

// =====================================================================
// Kernel 4: apply transport.  new_state[m,d] = N * sum_n T[n,m]*state[n,d],
//   T[n,m] = exp(w[n] + logb + (f[n]+g[m]-cost[n,m])/REG).
// Block = (batch, 16 m-columns); group j of 16 threads owns column m_j.
// =====================================================================
__global__ __launch_bounds__(256) void ot_apply(
    const float* __restrict__ state, const float* __restrict__ weight,
    const float* __restrict__ sx, const float* __restrict__ sq,
    const float* __restrict__ f, const float* __restrict__ g,
    float* __restrict__ out_state, float* __restrict__ out_logw)
{
  int b = blockIdx.x >> 7;
  int mbase = (blockIdx.x & 127) << 4;
  int grp = threadIdx.x >> 4;
  int k   = threadIdx.x & 15;
  int m = mbase + grp;

  __shared__ float sxm[16][DIM];
  __shared__ float accS[16][DIM];
  __shared__ float gsh[16], sqmsh[16];

  sxm[grp][k]  = sx[((size_t)(b * NP + m)) * DIM + k];
  accS[grp][k] = 0.f;
  if (k == 0) { gsh[grp] = g[b * NP + m]; sqmsh[grp] = sq[b * NP + m]; }
  __syncthreads();

  float sxmr[DIM];
#pragma unroll
  for (int d = 0; d < DIM; ++d) sxmr[d] = sxm[grp][d];
  float gm = gsh[grp], sqm = sqmsh[grp];

  float acc[DIM];
#pragma unroll
  for (int d = 0; d < DIM; ++d) acc[d] = 0.f;

  for (int n = k; n < NP; n += 16) {
    const float* sxr = sx + ((size_t)(b * NP + n)) * DIM;
    float dot = 0.f;
#pragma unroll
    for (int d = 0; d < DIM; ++d) dot = __builtin_fmaf(sxr[d], sxmr[d], dot);
    float cost = sq[b * NP + n] + sqm - 2.f * dot;
    float lt = weight[b * NP + n] + LOGB + (f[b * NP + n] + gm - cost) * INV_REG;
    float T = __expf(lt);
    const float* str = state + ((size_t)(b * NP + n)) * DIM;
#pragma unroll
    for (int d = 0; d < DIM; ++d) acc[d] = __builtin_fmaf(T, str[d], acc[d]);
  }

#pragma unroll
  for (int d = 0; d < DIM; ++d) atomicAdd(&accS[grp][d], acc[d]);
  __syncthreads();

  out_state[((size_t)(b * NP + m)) * DIM + k] = (float)NP * accS[grp][k];
  if (k == 0) out_logw[b * NP + m] = LOGB;
}

// =====================================================================
// Host-side launcher
// =====================================================================
extern "C" void kernel_launch(void* const* d_in, const int* in_sizes, int n_in,
                              void* d_out, int out_size, void* d_ws, size_t ws_size,
                              hipStream_t stream) {
  (void)in_sizes; (void)n_in; (void)out_size; (void)ws_size;
  const float* state  = (const float*)d_in[0];   // (B,N,D) f32
  const float* weight = (const float*)d_in[1];   // (B,N)   f32

  float* ws = (float*)d_ws;
  float* sx  = ws;                        // B*N*D = 131072
  float* sq  = sx + (size_t)NB * NP * DIM;   // +8192
  float* f   = sq + (size_t)NB * NP;         // +8192
  float* g   = f  + (size_t)NB * NP;         // +8192
  float* eps = g  + (size_t)NB * NP;         // +4
  int*   done = (int*)(eps + NB);            // +4
  unsigned int* upd = (unsigned int*)(done + NB);

  float* out_state = (float*)d_out;                       // B*N*D
  float* out_logw  = out_state + (size_t)NB * NP * DIM;   // B*N

  ot_prep<<<NB, 256, 0, stream>>>(state, sx, sq, f, g, eps, done, upd);

  for (int it = 0; it < MAX_IT; ++it) {
    // f-step: prior = log_b (constant), dual = g, writes f
    sinkhorn_half<<<64, 256, 0, stream>>>(sx, sq, g, f, nullptr, LOGB,
                                          eps, done, upd);
    // g-step: prior = weight, dual = f, writes g
    sinkhorn_half<<<64, 256, 0, stream>>>(sx, sq, f, g, weight, 0.f,
                                          eps, done, upd);
    sinkhorn_update<<<1, 32, 0, stream>>>(eps, done, upd);
  }

  ot_apply<<<NB * (NP / 16), 256, 0, stream>>>(state, weight, sx, sq, f, g,
                                               out_state, out_logw);
}